// FraudGNN_51264729645586
// MI455X (gfx1250) — compile-verified
//
#include <hip/hip_runtime.h>
#include <hip/hip_bf16.h>
#include <math.h>

typedef __attribute__((ext_vector_type(2))) float v2f;
typedef __attribute__((ext_vector_type(8))) float v8f;

#define NEG_SLOPE 0.2f

// ---------------------------------------------------------------------------
// Generic 32-bit fill (used to zero agg/den and max-keys; 0u == 0.0f == min key)
// ---------------------------------------------------------------------------
__global__ void fill_u32(unsigned* __restrict__ p, unsigned v, long n) {
  long i = (long)blockIdx.x * blockDim.x + threadIdx.x;
  if (i < n) p[i] = v;
}

// ---------------------------------------------------------------------------
// FP32 WMMA GEMM: Out[M,Ncol] = A[M,K] @ B[K,Ncol] (+bias, optional ReLU)
// Block = 256 threads = 8 waves; block computes a 128x16 output tile (8 M-tiles
// sharing one N-tile). The 16-column B panel (K x 16, <= 32 KB of the WGP's
// 320 KB LDS) is staged cooperatively in LDS as K-adjacent float2 pairs, so
// each lane's B operand for V_WMMA_F32_16X16X4_F32 is one 8-byte ds_load.
// A-operand layout (32-bit A 16x4): lane M = lane&15, K sub-offset 2*(lane>>4).
// C/D layout: row = r + 8*(lane>>4), col = lane&15.
// ---------------------------------------------------------------------------
__global__ void __launch_bounds__(256)
gemm_wmma(const float* __restrict__ A, const float* __restrict__ B,
          const float* __restrict__ bias, float* __restrict__ Out,
          int M, int K, int Ncol, int doRelu) {
  extern __shared__ v2f Bs[];                 // [K/2][16] : Bs[kp*16+n] = {B[2kp][n], B[2kp+1][n]}
  int tilesN = Ncol >> 4;
  int bm   = blockIdx.x / tilesN;             // 128-row block index
  int tn   = blockIdx.x - bm * tilesN;
  int col0 = tn << 4;

  // Cooperative stage of the B panel into LDS.
  int npairs = (K >> 1) << 4;                 // (K/2) * 16 float2 elements
  for (int i = threadIdx.x; i < npairs; i += 256) {
    int kp = i >> 4, n = i & 15;
    v2f v;
    v.x = B[(long)(2 * kp)     * Ncol + col0 + n];
    v.y = B[(long)(2 * kp + 1) * Ncol + col0 + n];
    Bs[i] = v;
  }
  __syncthreads();

  int tm = (bm << 3) + (int)(threadIdx.x >> 5);   // this wave's M tile
  if ((tm << 4) >= M) return;                 // wave-uniform exit: EXEC all-1s for WMMA

  int lane = threadIdx.x & 31;
  int half = lane >> 4;                       // 0 or 1
  int l15  = lane & 15;
  int row  = (tm << 4) + l15;
  int rowc = row < M ? row : (M - 1);         // clamp ragged M (loads only)
  int kh   = half << 1;                       // K sub-offset: 0 or 2

  v8f acc = {};
  const float* Arow = A + (long)rowc * K + kh;
  for (int k = 0; k < K; k += 4) {
    v2f a = *(const v2f*)(Arow + k);          // 8B-aligned global load
    v2f b = Bs[(((k + kh) >> 1) << 4) + l15]; // one ds_load_b64
    acc = __builtin_amdgcn_wmma_f32_16x16x4_f32(
        /*neg_a=*/false, a, /*neg_b=*/false, b,
        /*c_mod=*/(short)0, acc, /*reuse_a=*/false, /*reuse_b=*/false);
  }

  int orow = (tm << 4) + (half << 3);
  int col  = col0 + l15;
  float bv = bias ? bias[col] : 0.0f;
#pragma unroll
  for (int r = 0; r < 8; ++r) {
    if (orow + r < M) {
      float v = acc[r] + bv;
      if (doRelu) v = fmaxf(v, 0.0f);
      Out[(long)(orow + r) * Ncol + col] = v;
    }
  }
}

// ---------------------------------------------------------------------------
// Attention scores: s[n,h] = <h[n,h,:], a_src[h,:]>, d likewise.
// One wave per (node, head): lane-strided coalesced loads + shuffle reduction.
// ---------------------------------------------------------------------------
__global__ void __launch_bounds__(256)
att_scores(const float* __restrict__ h, const float* __restrict__ a_src,
           const float* __restrict__ a_dst, float* __restrict__ s,
           float* __restrict__ d, int NH, int H, int C) {
  int wave = (int)((blockIdx.x * blockDim.x + threadIdx.x) >> 5);
  int lane = threadIdx.x & 31;
  if (wave >= NH) return;
  int head = wave % H;
  const float* hp = h + (long)wave * C;
  const float* as = a_src + head * C;
  const float* ad = a_dst + head * C;
  float ss = 0.0f, dd = 0.0f;
  for (int c = lane; c < C; c += 32) {
    float hv = hp[c];
    ss += hv * as[c];
    dd += hv * ad[c];
  }
  for (int o = 16; o; o >>= 1) {
    ss += __shfl_down(ss, o, 32);
    dd += __shfl_down(dd, o, 32);
  }
  if (lane == 0) { s[wave] = ss; d[wave] = dd; }
}

// Order-preserving float<->uint key for atomic max (handles negatives; init key = 0u)
__device__ __forceinline__ unsigned fkey(float f) {
  unsigned b = __float_as_uint(f);
  return (b & 0x80000000u) ? ~b : (b | 0x80000000u);
}
__device__ __forceinline__ float funkey(unsigned k) {
  unsigned b = (k & 0x80000000u) ? (k ^ 0x80000000u) : ~k;
  return __uint_as_float(b);
}

// Pass 1: e = leaky_relu(s[src]+d[dst]); segment-max into mkey[dst]
__global__ void __launch_bounds__(256)
edge_max(const int* __restrict__ srcI, const int* __restrict__ dstI,
         const float* __restrict__ s, const float* __restrict__ d,
         float* __restrict__ ebuf, unsigned* __restrict__ mkey,
         int E, int Etot, int H) {
  int e = blockIdx.x * blockDim.x + threadIdx.x;
  if (e >= Etot) return;
  int sn = (e < E) ? srcI[e] : (e - E);    // self-loops appended after real edges
  int dn = (e < E) ? dstI[e] : (e - E);
  for (int h = 0; h < H; ++h) {
    float ev = s[sn * H + h] + d[dn * H + h];
    ev = (ev > 0.0f) ? ev : NEG_SLOPE * ev;
    ebuf[(long)e * H + h] = ev;
    atomicMax(&mkey[dn * H + h], fkey(ev));
  }
}

// Pass 2: ex = exp(e - max[dst]); segment-sum into den[dst]
__global__ void __launch_bounds__(256)
edge_exp(const int* __restrict__ dstI, float* __restrict__ ebuf,
         const unsigned* __restrict__ mkey, float* __restrict__ den,
         int E, int Etot, int H) {
  int e = blockIdx.x * blockDim.x + threadIdx.x;
  if (e >= Etot) return;
  int dn = (e < E) ? dstI[e] : (e - E);
  for (int h = 0; h < H; ++h) {
    float ex = __expf(ebuf[(long)e * H + h] - funkey(mkey[dn * H + h]));
    ebuf[(long)e * H + h] = ex;
    atomicAdd(&den[dn * H + h], ex);
  }
}

// Pass 3: agg[dst,h,:] += h[src,h,:] * (ex / den[dst,h]).  One wave per edge.
__global__ void __launch_bounds__(256)
edge_agg(const int* __restrict__ srcI, const int* __restrict__ dstI,
         const float* __restrict__ hbuf, const float* __restrict__ ebuf,
         const float* __restrict__ den, float* __restrict__ agg,
         int E, int Etot, int H, int C) {
  int wave = (int)((blockIdx.x * blockDim.x + threadIdx.x) >> 5);
  int lane = threadIdx.x & 31;
  if (wave >= Etot) return;
  int e = wave;
  int sn = (e < E) ? srcI[e] : (e - E);
  int dn = (e < E) ? dstI[e] : (e - E);
  for (int hd = 0; hd < H; ++hd) {
    float alpha = ebuf[(long)e * H + hd] / den[dn * H + hd];
    const float* hs = hbuf + ((long)sn * H + hd) * C;
    float* ap = agg + ((long)dn * H + hd) * C;
    for (int c = lane; c < C; c += 32)
      atomicAdd(&ap[c], hs[c] * alpha);
  }
}

// In-place bias + optional ReLU over [N, D]
__global__ void __launch_bounds__(256)
bias_act(float* __restrict__ out, const float* __restrict__ b,
         long total, int D, int doRelu) {
  long i = (long)blockIdx.x * blockDim.x + threadIdx.x;
  if (i >= total) return;
  float v = out[i] + b[i % D];
  out[i] = doRelu ? fmaxf(v, 0.0f) : v;
}

// Classifier head: out = log_softmax(hin[64] @ cW2[64,2] + cb2)
__global__ void __launch_bounds__(256)
cls_head(const float* __restrict__ hin, const float* __restrict__ W,
         const float* __restrict__ b, float* __restrict__ out, int N) {
  int n = blockIdx.x * blockDim.x + threadIdx.x;
  if (n >= N) return;
  const float* hp = hin + (long)n * 64;
  float o0 = b[0], o1 = b[1];
  for (int j = 0; j < 64; ++j) {
    float v = hp[j];
    o0 += v * W[j * 2 + 0];
    o1 += v * W[j * 2 + 1];
  }
  float mx = fmaxf(o0, o1);
  float l0 = o0 - mx, l1 = o1 - mx;
  float lse = __logf(__expf(l0) + __expf(l1));
  out[(long)n * 2 + 0] = l0 - lse;
  out[(long)n * 2 + 1] = l1 - lse;
}

// ---------------------------------------------------------------------------
extern "C" void kernel_launch(void* const* d_in, const int* in_sizes, int n_in,
                              void* d_out, int out_size, void* d_ws, size_t ws_size,
                              hipStream_t stream) {
  (void)n_in; (void)out_size; (void)ws_size;
  const float* x   = (const float*)d_in[0];
  const int*   ei  = (const int*)  d_in[1];
  const float* W1  = (const float*)d_in[2];
  const float* as1 = (const float*)d_in[3];
  const float* ad1 = (const float*)d_in[4];
  const float* b1  = (const float*)d_in[5];
  const float* W2  = (const float*)d_in[6];
  const float* as2 = (const float*)d_in[7];
  const float* ad2 = (const float*)d_in[8];
  const float* b2  = (const float*)d_in[9];
  const float* W3  = (const float*)d_in[10];
  const float* as3 = (const float*)d_in[11];
  const float* ad3 = (const float*)d_in[12];
  const float* b3  = (const float*)d_in[13];
  const float* cW1 = (const float*)d_in[14];
  const float* cb1 = (const float*)d_in[15];
  const float* cW2 = (const float*)d_in[16];
  const float* cb2 = (const float*)d_in[17];

  const int N    = in_sizes[0] / 64;   // F_in = 64
  const int E    = in_sizes[1] / 2;    // edge_index is [2, E]
  const int Etot = E + N;              // + self loops
  const int* srcI = ei;
  const int* dstI = ei + E;

  // ---- carve workspace ----
  char* ws = (char*)d_ws;
  size_t off = 0;
  auto carve = [&](size_t bytes) -> char* {
    char* p = ws + off;
    off += (bytes + 255) & ~(size_t)255;
    return p;
  };
  float*    bufP = (float*)carve((size_t)N * 512 * 4);  // layer activations / agg
  float*    bufQ = (float*)carve((size_t)N * 512 * 4);  // GEMM outputs (h)
  float*    sS   = (float*)carve((size_t)N * 4 * 4);
  float*    sD   = (float*)carve((size_t)N * 4 * 4);
  unsigned* mkey = (unsigned*)carve((size_t)N * 4 * 4);
  float*    den  = (float*)carve((size_t)N * 4 * 4);
  float*    ebuf = (float*)carve((size_t)Etot * 4 * 4);

  auto launch_gemm = [&](const float* Ain, const float* W, const float* bias,
                         float* outp, int M, int K, int D, int relu) {
    int tilesN = D / 16;
    int blocksM = (M + 127) / 128;
    size_t shmem = (size_t)K * 16 * sizeof(float);   // B panel (<=32KB)
    gemm_wmma<<<blocksM * tilesN, 256, shmem, stream>>>(Ain, W, bias, outp, M, K, D, relu);
  };

  // ---- one GAT layer ----
  auto gat = [&](const float* in, int K, const float* W, const float* as,
                 const float* ad, const float* b, int H, int C, int relu,
                 float* hbuf, float* aggbuf) {
    int D = H * C;
    launch_gemm(in, W, nullptr, hbuf, N, K, D, 0);
    int NH = N * H;
    att_scores<<<(NH + 7) / 8, 256, 0, stream>>>(hbuf, as, ad, sS, sD, NH, H, C);
    long aggN = (long)N * D;
    fill_u32<<<(int)((aggN + 255) / 256), 256, 0, stream>>>((unsigned*)aggbuf, 0u, aggN);
    fill_u32<<<(NH + 255) / 256, 256, 0, stream>>>(mkey, 0u, NH);
    fill_u32<<<(NH + 255) / 256, 256, 0, stream>>>((unsigned*)den, 0u, NH);
    edge_max<<<(Etot + 255) / 256, 256, 0, stream>>>(srcI, dstI, sS, sD, ebuf, mkey, E, Etot, H);
    edge_exp<<<(Etot + 255) / 256, 256, 0, stream>>>(dstI, ebuf, mkey, den, E, Etot, H);
    edge_agg<<<(Etot + 7) / 8, 256, 0, stream>>>(srcI, dstI, hbuf, ebuf, den, aggbuf, E, Etot, H, C);
    bias_act<<<(int)((aggN + 255) / 256), 256, 0, stream>>>(aggbuf, b, aggN, D, relu);
  };

  // Layer 1: x[N,64] -> [N,512], ReLU
  gat(x, 64, W1, as1, ad1, b1, 4, 128, 1, bufQ, bufP);
  // Layer 2: [N,512] -> [N,512], ReLU  (gemm consumes bufP before agg re-zeroes it)
  gat(bufP, 512, W2, as2, ad2, b2, 4, 128, 1, bufQ, bufP);
  // Layer 3: [N,512] -> [N,128], heads=1 (mean over 1 head = identity), no ReLU
  gat(bufP, 512, W3, as3, ad3, b3, 1, 128, 0, bufQ, bufP);

  // Classifier: relu(h3 @ cW1 + cb1) -> [N,64] via WMMA, then 64->2 + log_softmax
  launch_gemm(bufP, cW1, cb1, bufQ, N, 128, 64, 1);
  cls_head<<<(N + 255) / 256, 256, 0, stream>>>(bufQ, cW2, cb2, (float*)d_out, N);
}